// LearnableLogOptimalTransport_56831007261099
// MI455X (gfx1250) — compile-verified
//
#include <hip/hip_runtime.h>
#include <hip/hip_bf16.h>
#include <math.h>
#include <stdint.h>

// Sinkhorn-in-log-space for (32, 1025, 1025), iterated in-place in d_out.
// Row/Col passes: TDM tensor_load_to_lds stages a 16-line tile, per-lane f32
// partial reductions, one v_wmma_f32_16x16x4_f32 does the cross-lane sum.

#define BDIM 1025
#define NBATCH 32
#define TILE 16
#define RSTRIDE 1028   // padded LDS row stride (floats) for the row kernel
#define NTILES 65      // ceil(1025/16)

typedef float        v2f __attribute__((ext_vector_type(2)));
typedef float        v8f __attribute__((ext_vector_type(8)));
typedef unsigned int v4u __attribute__((ext_vector_type(4)));
typedef int          v4i __attribute__((ext_vector_type(4)));
typedef int          v8i __attribute__((ext_vector_type(8)));

// Cross-lane reduction via WMMA: lane l holds 2 f32 partials for line n=l&15,
// K-slots {2h, 2h+1} (h=l>>4).  D = ones(16x4) x partials(4x16):
// D[m][n] = sum_k B[k][n] = total for line n, replicated over m -> each lane's
// d[0] (M = 0 or 8, N = l&15) is its own line's total.
__device__ __forceinline__ float wmma_linesum(float s0, float s1) {
    v2f a; a.x = 1.0f; a.y = 1.0f;     // A = ones, 16x4 f32
    v2f b; b.x = s0;   b.y = s1;       // B = partial sums, 4x16 f32
    v8f c = {};
    c = __builtin_amdgcn_wmma_f32_16x16x4_f32(false, a, false, b,
                                              (short)0, c, false, false);
    return c[0];
}

__device__ __forceinline__ void tdm_load(v4u g0, v8i g1, v4i g2, v4i g3) {
    // clang-23 lane: 6-arg form — 4 descriptor groups + extra v8i + cpol.
    v8i gz; gz[0]=0; gz[1]=0; gz[2]=0; gz[3]=0; gz[4]=0; gz[5]=0; gz[6]=0; gz[7]=0;
    __builtin_amdgcn_tensor_load_to_lds(g0, g1, g2, g3, gz, 0);
    __builtin_amdgcn_s_wait_tensorcnt(0);
    asm volatile("" ::: "memory");     // compiler barrier: LDS now populated
}

__device__ __forceinline__ v4u tdm_group0(uint32_t lds_addr, const float* gptr) {
    uint64_t ga = (uint64_t)(uintptr_t)gptr;
    v4u g0;
    g0.x = 1u;                                     // count=1 (valid user D#)
    g0.y = lds_addr;                               // LDS byte address
    g0.z = (uint32_t)ga;                           // global_addr[31:0]
    g0.w = ((uint32_t)(ga >> 32) & 0x01FFFFFFu)    // global_addr[56:32]
         | (2u << 30);                             // type=2 ("image")
    return g0;
}

// ---------------------------------------------------------------------------
// Augment: Z[b,i,j] = cost[b,i,j] if i<1024 && j<1024 else alpha  (log domain)
// ---------------------------------------------------------------------------
__global__ void augment_kernel(const float* __restrict__ cost,
                               const float* __restrict__ alpha_p,
                               float* __restrict__ Z, long total) {
    const float alpha = alpha_p[0];
    const long plane = (long)BDIM * BDIM;
    for (long idx = (long)blockIdx.x * blockDim.x + threadIdx.x; idx < total;
         idx += (long)gridDim.x * blockDim.x) {
        long rem = idx % plane;
        int  b   = (int)(idx / plane);
        int  i   = (int)(rem / BDIM);
        int  j   = (int)(rem % BDIM);
        float v = (i < 1024 && j < 1024)
                    ? cost[((size_t)b * 1024 + i) * 1024 + j] : alpha;
        Z[idx] = v;
    }
}

// ---------------------------------------------------------------------------
// Row pass: logsumexp-normalize along the last axis.  One wave per 16-row tile.
// TDM iterate mode: 16 iterations of a 1025-float line, LDS stride 1028 floats.
// ---------------------------------------------------------------------------
__global__ __launch_bounds__(32) void row_pass(float* __restrict__ Z) {
    __shared__ float tile[TILE * RSTRIDE];

    const int b  = blockIdx.y;
    const int R0 = blockIdx.x * TILE;              // first row this tile OWNS
    const int Rs = min(R0, BDIM - TILE);           // shifted start (last tile)
    float* zb = Z + (size_t)b * BDIM * BDIM;

    // --- TDM descriptor: 16 x (1x1025) lines, iterate mode for LDS padding ---
    v4u g0 = tdm_group0((uint32_t)(uintptr_t)&tile[0], zb + (size_t)Rs * BDIM);
    v8i g1;
    g1[0] = (2 << 16) | (1 << 19);  // data_size=4B, iterate_enable=1
    g1[1] = (1025 << 16);           // tensor_dim0[15:0]=1025 (bits 63:48)
    g1[2] = (1 << 16);              // tensor_dim0 hi=0 | tensor_dim1[15:0]=1
    g1[3] = (1025 << 16);           // tensor_dim1 hi=0 | tile_dim0=1025
    g1[4] = 1;                      // tile_dim1=1, tile_dim2=0
    g1[5] = 1025;                   // tensor_dim0_stride lo32
    g1[6] = 0;                      // stride hi | tensor_dim1_stride lo
    g1[7] = 0;
    v4i g2;                         // iterate controls
    g2[0] = 0;                      // tensor_dim2
    g2[1] = RSTRIDE;                // lds_addr_increment   (data_size units)
    g2[2] = 1025;                   // global_addr_increment(data_size units)
    g2[3] = (15 << 16);             // iterate_count = 15 -> 16 lines
    v4i g3; g3[0] = 0; g3[1] = 0; g3[2] = 0; g3[3] = 0;
    tdm_load(g0, g1, g2, g3);

    const int lane = threadIdx.x;
    const int n = lane & 15;                       // row within tile
    const int h = lane >> 4;                       // K half
    const float* lrow = &tile[n * RSTRIDE];

    // pass 1: row max (lane covers cols {4j+2h, 4j+2h+1})
    float m = -__builtin_inff();
    #pragma unroll 4
    for (int j = 0; j < 256; ++j) {
        int c = 4 * j + 2 * h;
        m = fmaxf(m, fmaxf(lrow[c], lrow[c + 1]));
    }
    if (h == 0) m = fmaxf(m, lrow[1024]);          // tail column
    m = fmaxf(m, __shfl_xor(m, 16, 32));           // combine K halves

    // pass 2: partial sums of exp, then WMMA cross-lane reduce
    float s0 = 0.0f, s1 = 0.0f;
    #pragma unroll 4
    for (int j = 0; j < 256; ++j) {
        int c = 4 * j + 2 * h;
        s0 += __expf(lrow[c]     - m);
        s1 += __expf(lrow[c + 1] - m);
    }
    if (h == 0) s0 += __expf(lrow[1024] - m);
    const float lse = m + __logf(wmma_linesum(s0, s1));

    // pass 3: subtract & write (coalesced; skip rows owned by previous tile)
    for (int r = 0; r < TILE; ++r) {
        if (Rs + r < R0) continue;                 // uniform predicate
        const float lr = __shfl(lse, r, 32);       // lane r holds row r's LSE
        float* grow = zb + (size_t)(Rs + r) * BDIM;
        const float* srow = &tile[r * RSTRIDE];
        #pragma unroll
        for (int j = 0; j < 33; ++j) {
            int c = j * 32 + lane;
            if (c < BDIM) grow[c] = srow[c] - lr;
        }
    }
}

// ---------------------------------------------------------------------------
// Col pass: logsumexp-normalize along axis -2.  One wave per 16-column strip.
// TDM 2D tile: tile_dim0=16 cols (contiguous 64B lines), tile_dim1=1025 rows.
// ---------------------------------------------------------------------------
__global__ __launch_bounds__(32) void col_pass(float* __restrict__ Z,
                                               int apply_exp) {
    __shared__ float tile[BDIM * TILE];            // [row][col], stride 16

    const int b  = blockIdx.y;
    const int C0 = blockIdx.x * TILE;              // first col this strip OWNS
    const int Cs = min(C0, BDIM - TILE);
    float* zb = Z + (size_t)b * BDIM * BDIM;

    v4u g0 = tdm_group0((uint32_t)(uintptr_t)&tile[0], zb + Cs);
    v8i g1;
    g1[0] = (2 << 16);              // data_size=4B
    g1[1] = (16 << 16);             // tensor_dim0 = 16
    g1[2] = (1025 << 16);           // tensor_dim1[15:0] = 1025
    g1[3] = (16 << 16);             // tensor_dim1 hi=0 | tile_dim0=16
    g1[4] = 1025;                   // tile_dim1 = 1025, tile_dim2 = 0
    g1[5] = 1025;                   // tensor_dim0_stride = 1025
    g1[6] = 0;
    g1[7] = 0;
    v4i g2; g2[0] = 0; g2[1] = 0; g2[2] = 0; g2[3] = 0;
    v4i g3; g3[0] = 0; g3[1] = 0; g3[2] = 0; g3[3] = 0;
    tdm_load(g0, g1, g2, g3);

    const int lane = threadIdx.x;
    const int n = lane & 15;                       // column within strip
    const int h = lane >> 4;

    float m = -__builtin_inff();
    #pragma unroll 4
    for (int j = 0; j < 256; ++j) {
        int r = 4 * j + 2 * h;
        m = fmaxf(m, fmaxf(tile[r * TILE + n], tile[(r + 1) * TILE + n]));
    }
    if (h == 0) m = fmaxf(m, tile[1024 * TILE + n]);
    m = fmaxf(m, __shfl_xor(m, 16, 32));

    float s0 = 0.0f, s1 = 0.0f;
    #pragma unroll 4
    for (int j = 0; j < 256; ++j) {
        int r = 4 * j + 2 * h;
        s0 += __expf(tile[r * TILE + n]       - m);
        s1 += __expf(tile[(r + 1) * TILE + n] - m);
    }
    if (h == 0) s0 += __expf(tile[1024 * TILE + n] - m);
    const float lse = m + __logf(wmma_linesum(s0, s1));

    // write: lane owns column Cs+n; halves interleave rows (2j+h)
    const bool owned = (Cs + n) >= C0;
    for (int j = 0; j < 512; ++j) {
        int r = 2 * j + h;
        float v = tile[r * TILE + n] - lse;
        if (apply_exp) v = __expf(v);
        if (owned) zb[(size_t)r * BDIM + Cs + n] = v;
    }
    if (h == 0) {                                  // row 1024
        float v = tile[1024 * TILE + n] - lse;
        if (apply_exp) v = __expf(v);
        if (owned) zb[(size_t)1024 * BDIM + Cs + n] = v;
    }
}

// ---------------------------------------------------------------------------
extern "C" void kernel_launch(void* const* d_in, const int* in_sizes, int n_in,
                              void* d_out, int out_size, void* d_ws,
                              size_t ws_size, hipStream_t stream) {
    const float* cost  = (const float*)d_in[0];
    const float* alpha = (const float*)d_in[1];
    float* Z = (float*)d_out;                      // iterate in-place in d_out

    const long total = (long)NBATCH * BDIM * BDIM;
    augment_kernel<<<8192, 256, 0, stream>>>(cost, alpha, Z, total);

    dim3 grid(NTILES, NBATCH);
    for (int it = 0; it < 100; ++it) {
        row_pass<<<grid, 32, 0, stream>>>(Z);
        col_pass<<<grid, 32, 0, stream>>>(Z, (it == 99) ? 1 : 0);
    }
}